// Model_446676599203
// MI455X (gfx1250) — compile-verified
//
#include <hip/hip_runtime.h>
#include <hip/hip_bf16.h>

// ---------------------------------------------------------------------------
// Mamba forecaster on gfx1250 (CDNA5, wave32, WMMA f32 16x16x4, async->LDS,
// double-buffered weight staging).
// B=16, L=512, D_MODEL=128, D_INNER=256, D_STATE=16, D_CONV=4, DT_RANK=8,
// FORECAST=96.
// ---------------------------------------------------------------------------

#define BB 16
#define LL 512
#define DMODEL 128
#define DINNER 256
#define DSTATE 16
#define DTRANK 8
#define FORECAST 96
#define BL (BB * LL)            // 8192 rows
#define KFC (LL * DMODEL)       // 65536
#define FC_SPLIT 64
#define FC_KC (KFC / FC_SPLIT)  // 1024
#define NPADX 48                // x_proj N padded 40 -> 48
#define LDSB_STRIDE 34          // floats; even (b64 align) + bank spread

typedef __attribute__((ext_vector_type(2))) float v2f;
typedef __attribute__((ext_vector_type(8))) float v8f;

// ---- CDNA5 async global->LDS copy (ASYNCcnt) with safe fallback ------------
#if defined(__gfx1250__) &&                                         \
    __has_builtin(__builtin_amdgcn_global_load_async_to_lds_b64) && \
    __has_builtin(__builtin_amdgcn_s_wait_asynccnt)
#define USE_ASYNC_LDS 1
#else
#define USE_ASYNC_LDS 0
#endif

// Builtin signature (probe-confirmed via hipcc diagnostic): V2i AS1* src,
// V2i AS3* dst, imm offset, imm cpol.
typedef int v2i_gnu __attribute__((vector_size(8)));
typedef __attribute__((address_space(1))) v2i_gnu v2i_as1;
typedef __attribute__((address_space(3))) v2i_gnu v2i_as3;

__device__ __forceinline__ void copy_b64_to_lds(const float* g, float* l) {
#if USE_ASYNC_LDS
  __builtin_amdgcn_global_load_async_to_lds_b64(
      (v2i_as1*)(unsigned long long)g,
      (v2i_as3*)(unsigned int)(unsigned long long)l, 0, 0);
#else
  l[0] = g[0];
  l[1] = g[1];
#endif
}

__device__ __forceinline__ void wait_async_copies() {
#if USE_ASYNC_LDS
  __builtin_amdgcn_s_wait_asynccnt(0);
#endif
}

// ---------------------------------------------------------------------------
// fp32 WMMA GEMM, LDS-staged weights, compile-time K:
//   C[M,N] = A[M,K] * B[N,K]^T
// Block = 256 threads (8 waves) = 8 stacked M-tiles x 1 N-tile (128x16 of C).
// Weight chunks (16 x KC) double-buffered in LDS via ASYNCcnt copies; the
// next chunk's copy overlaps the current chunk's WMMAs (async loads complete
// in order, so s_wait_asynccnt(0) at iteration top drains the chunk we read).
// K and KC are template constants -> the chunk loop fully unrolls, all guards
// fold, and each chunk is: batched ds_load_b64 frags, then KC/4 back-to-back
// v_wmma_f32_16x16x4_f32 (XDL-friendly, EXEC all-ones throughout).
// ---------------------------------------------------------------------------
template <int K, int KC>
__global__ __launch_bounds__(256)
void k_gemm_wmma_lds(const float* __restrict__ A, const float* __restrict__ B,
                     float* __restrict__ C, int N, int lda, int ldb, int ldc) {
  constexpr int CH = K / KC;                // chunks (4, 8 or 1)
  __shared__ float ldsB[2][16 * LDSB_STRIDE];

  const int tid = threadIdx.x;
  const int wv = tid >> 5;                  // wave 0..7 -> M-tile within block
  const int lane = tid & 31;
  const int Nt = N >> 4;
  const int bn = blockIdx.x % Nt;           // N-tile
  const int bm = blockIdx.x / Nt;           // block-row (8 M-tiles)
  const int hl = lane >> 4;
  const int lm = lane & 15;

  const int tm = bm * 8 + wv;
  const float* __restrict__ Ap = A + (size_t)(tm * 16 + lm) * lda + hl * 2;

  // Copy-phase indices: 256 threads x one b64 covers a 16 x 32 chunk.
  const int crow = tid >> 4;                // 0..15 weight row within tile
  const int ccol = (tid & 15) * 2;          // 0..30 (K offset, even)
  const float* __restrict__ Bp = B + (size_t)(bn * 16 + crow) * ldb + ccol;
  const bool cdo = (ccol < KC);             // compile-time when KC==32
  float* const ldst[2] = {&ldsB[0][crow * LDSB_STRIDE + ccol],
                          &ldsB[1][crow * LDSB_STRIDE + ccol]};

  if (cdo) copy_b64_to_lds(Bp, ldst[0]);    // prologue: chunk 0 -> buffer 0

  v8f acc = {};
#pragma unroll
  for (int c = 0; c < CH; ++c) {
    const int k0 = c * KC;

    // A fragments for this chunk (global b64s, independent of LDS state).
    v2f afr[KC / 4];
#pragma unroll
    for (int i = 0; i < KC / 4; ++i) {
      afr[i].x = Ap[k0 + i * 4];
      afr[i].y = Ap[k0 + i * 4 + 1];
    }
    if (c + 1 < CH) __builtin_prefetch(Ap + k0 + KC, 0, 0);

    wait_async_copies();                    // own chunk-c copy complete
    __syncthreads();                        // everyone's chunk-c data visible

    if (c + 1 < CH && cdo)                  // overlap next copy with WMMAs
      copy_b64_to_lds(Bp + k0 + KC, ldst[(c + 1) & 1]);

    // B fragments from LDS (batched ds_load_b64s, single wait).
    const float* __restrict__ blp = &ldsB[c & 1][lm * LDSB_STRIDE + hl * 2];
    v2f bfr[KC / 4];
#pragma unroll
    for (int i = 0; i < KC / 4; ++i) {
      bfr[i].x = blp[i * 4];
      bfr[i].y = blp[i * 4 + 1];
    }

#pragma unroll
    for (int i = 0; i < KC / 4; ++i)
      acc = __builtin_amdgcn_wmma_f32_16x16x4_f32(
          false, afr[i], false, bfr[i], (short)0, acc, false, false);

    __syncthreads();                        // reads done before buffer reuse
  }

  const int col = bn * 16 + lm;
#pragma unroll
  for (int r = 0; r < 8; ++r)
    C[(size_t)(tm * 16 + hl * 8 + r) * ldc + col] = acc[r];
}

// ---------------------------------------------------------------------------
// Pad x_proj_w (40,256) -> (48,256) with zero rows so its GEMM is guard-free.
// ---------------------------------------------------------------------------
__global__ __launch_bounds__(256)
void k_pad_xproj(const float* __restrict__ w, float* __restrict__ wp) {
  const int idx = blockIdx.x * blockDim.x + threadIdx.x;
  if (idx >= NPADX * DINNER) return;
  const int row = idx / DINNER;
  wp[idx] = (row < DTRANK + 2 * DSTATE) ? w[idx] : 0.f;
}

// ---------------------------------------------------------------------------
// Depthwise causal conv (k=4, left pad 3) + bias + SiLU.
// xz: (BL, 512) with xin in cols [0,256).  Output xact: (BL, 256).
// ---------------------------------------------------------------------------
__global__ __launch_bounds__(256)
void k_conv_silu(const float* __restrict__ xz, const float* __restrict__ conv_w,
                 const float* __restrict__ conv_b, float* __restrict__ xact) {
  const int idx = blockIdx.x * blockDim.x + threadIdx.x;
  if (idx >= BL * DINNER) return;
  const int d = idx & (DINNER - 1);
  const int bl = idx >> 8;                 // b*512 + l
  const int l = bl & (LL - 1);
  float acc = conv_b[d];
#pragma unroll
  for (int j = 0; j < 4; ++j) {
    const int ls = l - 3 + j;
    if (ls >= 0)
      acc += conv_w[d * 4 + j] * xz[(size_t)(bl - l + ls) * (2 * DINNER) + d];
  }
  xact[(size_t)bl * DINNER + d] = acc / (1.f + __expf(-acc));  // SiLU
}

// ---------------------------------------------------------------------------
// Selective-scan: one thread per (b,d), h[16] in VGPRs, L=512 steps.
// Fuses softplus(dt), exp recurrence, C-readout, D-skip and the z-gate.
// dbc row layout: [dt(8) | B(16) | C(16)], ld = 48.
// ---------------------------------------------------------------------------
__global__ __launch_bounds__(256)
void k_scan(const float* __restrict__ dbc, const float* __restrict__ dtlin,
            const float* __restrict__ xact, const float* __restrict__ xz,
            const float* __restrict__ dt_b, const float* __restrict__ A_log,
            const float* __restrict__ Dp, float* __restrict__ y) {
  const int idx = blockIdx.x * blockDim.x + threadIdx.x;
  if (idx >= BB * DINNER) return;
  const int d = idx & (DINNER - 1);
  const int b = idx >> 8;

  float An[DSTATE], h[DSTATE];
#pragma unroll
  for (int n = 0; n < DSTATE; ++n) {
    An[n] = -__expf(A_log[d * DSTATE + n]);
    h[n] = 0.f;
  }
  const float bias = dt_b[d];
  const float Dd = Dp[d];

  for (int l = 0; l < LL; ++l) {
    const size_t bl = (size_t)b * LL + l;
    const float xv = xact[bl * DINNER + d];
    const float xdt = dtlin[bl * DINNER + d] + bias;
    const float dt = (xdt > 20.f) ? xdt : __logf(1.f + __expf(xdt));
    const float* __restrict__ bc = dbc + bl * NPADX;
    float acc = xv * Dd;
#pragma unroll
    for (int n = 0; n < DSTATE; ++n) {
      const float dA = __expf(dt * An[n]);
      h[n] = dA * h[n] + (dt * bc[DTRANK + n]) * xv;   // B-input
      acc += h[n] * bc[DTRANK + DSTATE + n];           // C-readout
    }
    const float zv = xz[bl * (2 * DINNER) + DINNER + d];
    y[bl * DINNER + d] = acc * (zv / (1.f + __expf(-zv)));  // * SiLU(z)
  }
}

// ---------------------------------------------------------------------------
// fc1 split-K WMMA: out(16,96) = H(16,65536) * W(96,65536)^T.
// 32-wide K steps: 16 batched global b64 loads then 8 back-to-back WMMAs.
// ---------------------------------------------------------------------------
__global__ __launch_bounds__(256)
void k_fc1_partial(const float* __restrict__ H, const float* __restrict__ W,
                   float* __restrict__ part) {
  const int wave = (blockIdx.x * blockDim.x + threadIdx.x) >> 5;
  const int lane = threadIdx.x & 31;
  const int Nt = FORECAST / 16;             // 6
  if (wave >= Nt * FC_SPLIT) return;        // wave-uniform
  const int tn = wave % Nt;
  const int chunk = wave / Nt;
  const int hl = lane >> 4;
  const int lm = lane & 15;
  const int k0base = chunk * FC_KC;

  const float* __restrict__ Ap = H + (size_t)lm * KFC + k0base + hl * 2;
  const float* __restrict__ Bp = W + (size_t)(tn * 16 + lm) * KFC + k0base + hl * 2;

  v8f acc = {};
  for (int k0 = 0; k0 < FC_KC; k0 += 32) {
    v2f a[8], b[8];
#pragma unroll
    for (int i = 0; i < 8; ++i) {
      a[i].x = Ap[k0 + i * 4];
      a[i].y = Ap[k0 + i * 4 + 1];
      b[i].x = Bp[k0 + i * 4];
      b[i].y = Bp[k0 + i * 4 + 1];
    }
#pragma unroll
    for (int i = 0; i < 8; ++i)
      acc = __builtin_amdgcn_wmma_f32_16x16x4_f32(
          false, a[i], false, b[i], (short)0, acc, false, false);
  }

  const int col = tn * 16 + lm;
#pragma unroll
  for (int r = 0; r < 8; ++r)
    part[((size_t)chunk * BB + hl * 8 + r) * FORECAST + col] = acc[r];
}

__global__ __launch_bounds__(256)
void k_fc1_reduce(const float* __restrict__ part, const float* __restrict__ bias,
                  float* __restrict__ out) {
  const int idx = blockIdx.x * blockDim.x + threadIdx.x;
  if (idx >= BB * FORECAST) return;
  const int col = idx % FORECAST;
  const int row = idx / FORECAST;
  float s = bias[col];
  for (int c = 0; c < FC_SPLIT; ++c)
    s += part[((size_t)c * BB + row) * FORECAST + col];
  out[idx] = s;
}

// ---------------------------------------------------------------------------
// Host-side orchestration
// ---------------------------------------------------------------------------
extern "C" void kernel_launch(void* const* d_in, const int* in_sizes, int n_in,
                              void* d_out, int out_size, void* d_ws, size_t ws_size,
                              hipStream_t stream) {
  (void)in_sizes; (void)n_in; (void)out_size; (void)ws_size;

  const float* x          = (const float*)d_in[0];   // (B,L,128)
  const float* in_proj_w  = (const float*)d_in[1];   // (512,128)
  const float* conv_w     = (const float*)d_in[2];   // (256,4)
  const float* conv_b     = (const float*)d_in[3];   // (256)
  const float* x_proj_w   = (const float*)d_in[4];   // (40,256)
  const float* dt_proj_w  = (const float*)d_in[5];   // (256,8)
  const float* dt_proj_b  = (const float*)d_in[6];   // (256)
  const float* A_log      = (const float*)d_in[7];   // (256,16)
  const float* D_param    = (const float*)d_in[8];   // (256)
  const float* out_proj_w = (const float*)d_in[9];   // (128,256)
  const float* fc1_w      = (const float*)d_in[10];  // (96,65536)
  const float* fc1_b      = (const float*)d_in[11];  // (96)
  float* out = (float*)d_out;

  // Workspace layout (floats)
  float* w = (float*)d_ws;
  float* xz     = w;                                   // (BL,512)
  float* xact   = xz     + (size_t)BL * 512;           // (BL,256)
  float* dbc    = xact   + (size_t)BL * DINNER;        // (BL,48)
  float* dtlin  = dbc    + (size_t)BL * NPADX;         // (BL,256)
  float* yb     = dtlin  + (size_t)BL * DINNER;        // (BL,256)
  float* hmodel = yb     + (size_t)BL * DINNER;        // (BL,128)
  float* fc1p   = hmodel + (size_t)BL * DMODEL;        // (64,16,96)
  float* wpad   = fc1p   + (size_t)FC_SPLIT * BB * FORECAST;  // (48,256)

  auto gemm_grid = [](int M, int N) {                  // 128x16 of C per block
    return (M / 128) * (N / 16);
  };

  // 0) pad x_proj weights 40x256 -> 48x256 (zero rows 40..47)
  k_pad_xproj<<<(NPADX * DINNER + 255) / 256, 256, 0, stream>>>(x_proj_w, wpad);

  // 1) in_proj: xz = x @ in_proj_w^T   (8192x512, K=128)
  k_gemm_wmma_lds<DMODEL, 32><<<gemm_grid(BL, 2 * DINNER), 256, 0, stream>>>(
      x, in_proj_w, xz, 2 * DINNER, DMODEL, DMODEL, 2 * DINNER);

  // 2) depthwise causal conv + bias + SiLU -> xact
  k_conv_silu<<<(BL * DINNER + 255) / 256, 256, 0, stream>>>(
      xz, conv_w, conv_b, xact);

  // 3) x_proj: dbc = xact @ wpad^T  (8192x48, K=256)
  k_gemm_wmma_lds<DINNER, 32><<<gemm_grid(BL, NPADX), 256, 0, stream>>>(
      xact, wpad, dbc, NPADX, DINNER, DINNER, NPADX);

  // 4) dt_proj: dtlin = dbc[:, :8] @ dt_proj_w^T  (8192x256, K=8)
  k_gemm_wmma_lds<DTRANK, 8><<<gemm_grid(BL, DINNER), 256, 0, stream>>>(
      dbc, dt_proj_w, dtlin, DINNER, NPADX, DTRANK, DINNER);

  // 5) selective scan (fused softplus, recurrence, C-readout, D-skip, z-gate)
  k_scan<<<(BB * DINNER + 255) / 256, 256, 0, stream>>>(
      dbc, dtlin, xact, xz, dt_proj_b, A_log, D_param, yb);

  // 6) out_proj: hmodel = y @ out_proj_w^T  (8192x128, K=256)
  k_gemm_wmma_lds<DINNER, 32><<<gemm_grid(BL, DMODEL), 256, 0, stream>>>(
      yb, out_proj_w, hmodel, DMODEL, DINNER, DINNER, DMODEL);

  // 7) fc1 split-K + deterministic reduce (+bias)
  {
    const int waves = (FORECAST / 16) * FC_SPLIT;      // 384
    k_fc1_partial<<<(waves + 7) / 8, 256, 0, stream>>>(hmodel, fc1_w, fc1p);
    k_fc1_reduce<<<(BB * FORECAST + 255) / 256, 256, 0, stream>>>(
        fc1p, fc1_b, out);
  }
}